// EdgeSoftmaxNodeFlow_23519240912985
// MI455X (gfx1250) — compile-verified
//
#include <hip/hip_runtime.h>
#include <hip/hip_bf16.h>

// Edge softmax over a graph (segmented max -> exp -> segmented sum).
// HBM-bound scatter/gather workload: no WMMA applies. CDNA5 paths used:
//   - native fp atomics via inline asm: global_atomic_max_num_f32 /
//     global_atomic_add_f32 at device scope (segmented reductions in L2)
//   - non-temporal stores for the streaming scores output so the 102 MB
//     logits array stays resident in the 192 MB L2 between passes
//   - wave32, b128 vector memory ops

#define TPB 256

typedef float v4f __attribute__((ext_vector_type(4)));

// Device-scope non-returning float atomics (CDNA5 VMEM ISA ops).
__device__ __forceinline__ void atomic_fmax_dev(float* p, float v) {
    asm volatile("global_atomic_max_num_f32 %0, %1, off scope:SCOPE_DEV"
                 :: "v"(p), "v"(v) : "memory");
}
__device__ __forceinline__ void atomic_fadd_dev(float* p, float v) {
    asm volatile("global_atomic_add_f32 %0, %1, off scope:SCOPE_DEV"
                 :: "v"(p), "v"(v) : "memory");
}

// ---------------------------------------------------------------------------
// Pass 0: init node-max workspace to -inf, zero the normalizer output region.
// ---------------------------------------------------------------------------
__global__ __launch_bounds__(TPB) void esm_init_kernel(float* __restrict__ maxbuf,
                                                       float* __restrict__ norm,
                                                       int nh) {
    int i = blockIdx.x * TPB + threadIdx.x;
    if (i < nh) {
        maxbuf[i] = -__builtin_inff();
        norm[i]   = 0.0f;
    }
}

// ---------------------------------------------------------------------------
// Pass 1: per-edge atomic max of H=8 head logits into node-max buffer.
// One thread per edge: 1x b32 index load + 2x b128 logit loads (RT hint:
// keep logits resident in L2 for pass 2) + 8 fp-max atomics.
// ---------------------------------------------------------------------------
__global__ __launch_bounds__(TPB) void esm_max8_kernel(const float* __restrict__ logits,
                                                       const int* __restrict__ dst,
                                                       float* __restrict__ maxbuf,
                                                       int E) {
    int e = blockIdx.x * TPB + threadIdx.x;
    if (e >= E) return;
    int d = dst[e];
    const v4f* l4 = reinterpret_cast<const v4f*>(logits + (size_t)e * 8);
    v4f a = l4[0];
    v4f b = l4[1];
    float* base = maxbuf + (size_t)d * 8;
    atomic_fmax_dev(base + 0, a.x);
    atomic_fmax_dev(base + 1, a.y);
    atomic_fmax_dev(base + 2, a.z);
    atomic_fmax_dev(base + 3, a.w);
    atomic_fmax_dev(base + 4, b.x);
    atomic_fmax_dev(base + 5, b.y);
    atomic_fmax_dev(base + 6, b.z);
    atomic_fmax_dev(base + 7, b.w);
}

// ---------------------------------------------------------------------------
// Pass 2: scores = exp(logit - max[dst]); atomic-add scores into normalizer.
// Logits re-read with NT hint (last use, likely L2-resident from pass 1);
// scores written with NT stores (never re-read -> don't evict logits from L2).
// ---------------------------------------------------------------------------
__global__ __launch_bounds__(TPB) void esm_score8_kernel(const float* __restrict__ logits,
                                                         const int* __restrict__ dst,
                                                         const float* __restrict__ maxbuf,
                                                         float* __restrict__ scores,
                                                         float* __restrict__ norm,
                                                         int E) {
    int e = blockIdx.x * TPB + threadIdx.x;
    if (e >= E) return;
    int d = dst[e];
    const v4f* l4 = reinterpret_cast<const v4f*>(logits + (size_t)e * 8);
    const v4f* m4 = reinterpret_cast<const v4f*>(maxbuf + (size_t)d * 8);
    v4f a = __builtin_nontemporal_load(l4);
    v4f b = __builtin_nontemporal_load(l4 + 1);
    v4f ma = m4[0];
    v4f mb = m4[1];
    v4f sa, sb;
    sa.x = __expf(a.x - ma.x);
    sa.y = __expf(a.y - ma.y);
    sa.z = __expf(a.z - ma.z);
    sa.w = __expf(a.w - ma.w);
    sb.x = __expf(b.x - mb.x);
    sb.y = __expf(b.y - mb.y);
    sb.z = __expf(b.z - mb.z);
    sb.w = __expf(b.w - mb.w);
    v4f* o4 = reinterpret_cast<v4f*>(scores + (size_t)e * 8);
    __builtin_nontemporal_store(sa, o4);
    __builtin_nontemporal_store(sb, o4 + 1);
    float* nb = norm + (size_t)d * 8;
    atomic_fadd_dev(nb + 0, sa.x);
    atomic_fadd_dev(nb + 1, sa.y);
    atomic_fadd_dev(nb + 2, sa.z);
    atomic_fadd_dev(nb + 3, sa.w);
    atomic_fadd_dev(nb + 4, sb.x);
    atomic_fadd_dev(nb + 5, sb.y);
    atomic_fadd_dev(nb + 6, sb.z);
    atomic_fadd_dev(nb + 7, sb.w);
}

// ---------------------------------------------------------------------------
// Generic-H fallbacks (scalar per (edge,head) element), in case H != 8.
// ---------------------------------------------------------------------------
__global__ __launch_bounds__(TPB) void esm_max_gen_kernel(const float* __restrict__ logits,
                                                          const int* __restrict__ dst,
                                                          float* __restrict__ maxbuf,
                                                          int E, int H) {
    int i = blockIdx.x * TPB + threadIdx.x;
    if (i >= E * H) return;
    int e = i / H;
    int h = i - e * H;
    int d = dst[e];
    atomic_fmax_dev(maxbuf + (size_t)d * H + h, logits[i]);
}

__global__ __launch_bounds__(TPB) void esm_score_gen_kernel(const float* __restrict__ logits,
                                                            const int* __restrict__ dst,
                                                            const float* __restrict__ maxbuf,
                                                            float* __restrict__ scores,
                                                            float* __restrict__ norm,
                                                            int E, int H) {
    int i = blockIdx.x * TPB + threadIdx.x;
    if (i >= E * H) return;
    int e = i / H;
    int h = i - e * H;
    int d = dst[e];
    float s = __expf(logits[i] - maxbuf[(size_t)d * H + h]);
    scores[i] = s;
    atomic_fadd_dev(norm + (size_t)d * H + h, s);
}

extern "C" void kernel_launch(void* const* d_in, const int* in_sizes, int n_in,
                              void* d_out, int out_size, void* d_ws, size_t ws_size,
                              hipStream_t stream) {
    const float* logits = (const float*)d_in[0];
    const int*   dst    = (const int*)d_in[1];

    const int EH = in_sizes[0];        // E * H
    const int E  = in_sizes[1];        // edges
    const int H  = EH / E;             // heads (8)
    const int NH = out_size - EH;      // N * H (normalizer element count)

    float* scores = (float*)d_out;          // [E*H]
    float* norm   = (float*)d_out + EH;     // [N*H]
    float* maxbuf = (float*)d_ws;           // [N*H] node-max scratch

    // Pass 0: init
    int gInit = (NH + TPB - 1) / TPB;
    esm_init_kernel<<<gInit, TPB, 0, stream>>>(maxbuf, norm, NH);

    if (H == 8) {
        int gE = (E + TPB - 1) / TPB;
        esm_max8_kernel<<<gE, TPB, 0, stream>>>(logits, dst, maxbuf, E);
        esm_score8_kernel<<<gE, TPB, 0, stream>>>(logits, dst, maxbuf, scores, norm, E);
    } else {
        int gEH = (EH + TPB - 1) / TPB;
        esm_max_gen_kernel<<<gEH, TPB, 0, stream>>>(logits, dst, maxbuf, E, H);
        esm_score_gen_kernel<<<gEH, TPB, 0, stream>>>(logits, dst, maxbuf, scores, norm, E, H);
    }
}